// MessagePassingBlock_67997922230575
// MI455X (gfx1250) — compile-verified
//
#include <hip/hip_runtime.h>
#include <hip/hip_bf16.h>

// ---------------------------------------------------------------------------
// MessagePassing: out[q] += w_e * (ref_feat[e_ref] @ kernel[e_kernel])
// Counting-sort edges by kernel index into 16-aligned buckets; one wave per
// 16-edge tile computes A(16x64) @ B(64x64) with fp32 WMMA (16x16x4) and
// atomically scatters rows into out. Kernel weights are pre-packed into WMMA
// B-fragment order so each lane loads its fragments as contiguous b128s.
// ---------------------------------------------------------------------------

typedef float v2f __attribute__((ext_vector_type(2)));
typedef float v8f __attribute__((ext_vector_type(8)));

#define NK 27           // number of kernels
#define D  64           // D1 == D2 == 64

// ws layout:
//   u32[0..26]   : counts -> bucket start offsets (after scan)
//   u32[32..58]  : scatter cursors
//   u32[64]      : tileCount
//   f32[128 .. 128+NK*4096)            : packedB (B fragments, 432 KB)
//   i32[128+NK*4096 .. +E+NK*16)       : bucket array of padded edge ids
#define PACKB_OFF 128
#define BUCKET_OFF (PACKB_OFF + NK * 4096)

// ---- init: zero out + counters, fill bucket with -1, histogram kernels ----
__global__ void mp_init(const int* __restrict__ eKernel,
                        unsigned* __restrict__ ws,
                        int* __restrict__ bucket,
                        float* __restrict__ out,
                        int E, int padSlots, int outN) {
    int i = blockIdx.x * blockDim.x + threadIdx.x;
    if (i < 128) ws[i] = 0u;
    if (i < padSlots) bucket[i] = -1;
    if (i < outN) out[i] = 0.0f;
    if (i < E) atomicAdd(&ws[eKernel[i]], 1u);
}

// ---- pack: reorder kernel weights into WMMA B-fragment order ----
// packedB[((k*4 + nt)*32 + lane)*32 + kb*2 + c] = B_k[kb*4 + hi*2 + c][nt*16 + m]
// with lane = hi*16 + m. Each lane's 16 kb-fragments are contiguous (128 B).
__global__ void mp_pack(const float* __restrict__ kernelW,
                        float* __restrict__ packedB) {
    int i = blockIdx.x * blockDim.x + threadIdx.x;
    if (i < NK * 4096) {
        int c    = i & 1;
        int kb   = (i >> 1) & 15;
        int lane = (i >> 5) & 31;
        int nt   = (i >> 10) & 3;
        int k    = i >> 12;
        int hi = lane >> 4, m = lane & 15;
        int kr = kb * 4 + hi * 2 + c;
        packedB[i] = kernelW[k * 4096 + kr * D + nt * 16 + m];
    }
}

// ---- scan: 16-aligned exclusive prefix over 27 buckets (single thread) ----
__global__ void mp_scan(unsigned* __restrict__ ws) {
    if (threadIdx.x == 0 && blockIdx.x == 0) {
        unsigned pos = 0;
        for (int k = 0; k < NK; ++k) {
            unsigned c = ws[k];
            ws[k]      = pos;   // bucket start
            ws[32 + k] = pos;   // scatter cursor
            pos += (c + 15u) & ~15u;
        }
        ws[64] = pos >> 4;      // total tiles
    }
}

// ---- scatter: place edge ids into their kernel's bucket ----
__global__ void mp_scatter(const int* __restrict__ eKernel,
                           unsigned* __restrict__ ws,
                           int* __restrict__ bucket, int E) {
    int i = blockIdx.x * blockDim.x + threadIdx.x;
    if (i < E) {
        int k = eKernel[i];
        unsigned pos = atomicAdd(&ws[32 + k], 1u);
        bucket[pos] = i;
    }
}

// ---- main: one wave per 16-edge tile; A(16x64) @ B(64x64) via fp32 WMMA ----
__global__ __launch_bounds__(256) void mp_main(
    const float* __restrict__ packedB,   // [27][4][32][32] B fragments
    const float* __restrict__ ref,       // [N,64]
    const int*   __restrict__ eKernel,
    const int*   __restrict__ eRef,
    const int*   __restrict__ eQuery,
    const float* __restrict__ eWeight,
    const int*   __restrict__ bucket,    // padded edge ids (-1 = pad)
    const unsigned* __restrict__ meta,   // meta[0] = tileCount
    float* __restrict__ out)             // [M,64]
{
    const int lane = threadIdx.x & 31;
    const int wave = threadIdx.x >> 5;
    const unsigned tile = blockIdx.x * 8u + wave;   // wave-uniform
    if (tile >= meta[0]) return;                    // uniform exit; EXEC stays full

    const int m  = lane & 15;   // matrix row handled by this lane (A side)
    const int hi = lane >> 4;   // half-wave: K-pair select for A/B fragments
    const long base = (long)tile * 16;

    // Per-row edge data (lane m and lane m+16 both hold row m's edge)
    int   eid = bucket[base + m];
    int   r   = (eid >= 0) ? eRef[eid]    : 0;
    int   q   = (eid >= 0) ? eQuery[eid]  : 0;
    float w   = (eid >= 0) ? eWeight[eid] : 0.0f;

    // Kernel index from slot 0 (always a valid edge for any live tile)
    int kIdx = __builtin_amdgcn_readfirstlane(eKernel[bucket[base]]);

    // Preload A fragments: a[kb] holds A[m][kb*4 + hi*2 + {0,1}], pre-scaled by w.
    // ISA 16x4 f32 A layout: VGPR0 = K0 (lanes<16) / K2 (lanes>=16), VGPR1 = K1/K3.
    v2f a[16];
    const float* rrow = ref + (long)r * D + hi * 2;
    #pragma unroll
    for (int kb = 0; kb < 16; ++kb) {
        float2 t = *(const float2*)(rrow + kb * 4);
        a[kb].x = t.x * w;
        a[kb].y = t.y * w;
    }

    // Hoisted scatter metadata: output row base + validity per C register j.
    // C layout: VGPR j -> row j (lanes<16) / row j+8 (lanes>=16); col = m.
    long  outBase[8];
    bool  rowValid[8];
    #pragma unroll
    for (int j = 0; j < 8; ++j) {
        const int row = j + hi * 8;
        const int qj = __shfl(q,   row, 32);
        const int ej = __shfl(eid, row, 32);
        rowValid[j] = (ej >= 0);
        outBase[j]  = (long)qj * D + m;
    }

    #pragma unroll
    for (int nt = 0; nt < 4; ++nt) {        // 4 column tiles of 16 (D2 = 64)
        // This lane's 16 contiguous B fragments for column tile nt (128 B).
        const float* bp = packedB + (((long)kIdx * 4 + nt) * 32 + lane) * 32;
        float4 bb[8];
        #pragma unroll
        for (int i = 0; i < 8; ++i) bb[i] = *(const float4*)(bp + i * 4);

        v8f c = {};
        #pragma unroll
        for (int kb = 0; kb < 16; ++kb) {
            v2f b;
            if (kb & 1) { b.x = bb[kb >> 1].z; b.y = bb[kb >> 1].w; }
            else        { b.x = bb[kb >> 1].x; b.y = bb[kb >> 1].y; }
            c = __builtin_amdgcn_wmma_f32_16x16x4_f32(
                    /*neg_a=*/false, a[kb],
                    /*neg_b=*/false, b,
                    /*c_mod=*/(short)0, c,
                    /*reuse_a=*/false, /*reuse_b=*/false);
        }
        #pragma unroll
        for (int j = 0; j < 8; ++j) {
            if (rowValid[j]) {
                atomicAdd(out + outBase[j] + nt * 16, c[j]);
            }
        }
    }
}

extern "C" void kernel_launch(void* const* d_in, const int* in_sizes, int n_in,
                              void* d_out, int out_size, void* d_ws, size_t ws_size,
                              hipStream_t stream) {
    const float* kernelW = (const float*)d_in[0];   // [27,64,64]
    const float* ref     = (const float*)d_in[1];   // [N,64]
    const int*   eKernel = (const int*)  d_in[2];   // [E]
    const int*   eRef    = (const int*)  d_in[3];   // [E]
    const int*   eQuery  = (const int*)  d_in[4];   // [E]
    // d_in[5] = num_queries (device scalar; M = out_size / 64 instead)
    const float* eWeight = (const float*)d_in[6];   // [E]
    float* out = (float*)d_out;

    const int E        = in_sizes[2];
    const int outN     = out_size;            // M * 64
    const int padSlots = E + NK * 16;

    unsigned* ws   = (unsigned*)d_ws;
    float* packedB = (float*)ws + PACKB_OFF;
    int*   bucket  = (int*)ws + BUCKET_OFF;

    int initN = outN > padSlots ? outN : padSlots;
    if (initN < 128) initN = 128;
    mp_init<<<(initN + 255) / 256, 256, 0, stream>>>(eKernel, ws, bucket, out,
                                                     E, padSlots, outN);
    mp_pack<<<(NK * 4096 + 255) / 256, 256, 0, stream>>>(kernelW, packedB);
    mp_scan<<<1, 32, 0, stream>>>(ws);
    mp_scatter<<<(E + 255) / 256, 256, 0, stream>>>(eKernel, ws, bucket, E);

    const int maxTiles  = (E + 15) / 16 + NK;       // upper bound on tileCount
    const int numBlocks = (maxTiles + 7) / 8;       // 8 waves (tiles) per block
    mp_main<<<numBlocks, 256, 0, stream>>>(packedB, ref, eKernel, eRef, eQuery,
                                           eWeight, bucket, ws + 64, out);
}